// GeometricFeatureExtraction_73100343378212
// MI455X (gfx1250) — compile-verified
//
#include <hip/hip_runtime.h>
#include <math.h>

namespace {

constexpr int BATCH = 2;
constexpr int C     = 128;
constexpr int N     = 8192;
constexpr int KNB   = 16;   // k neighbors
constexpr int CH    = 16;
constexpr float EPSF   = 1e-8f;
constexpr float BN_EPS = 1e-5f;

typedef __attribute__((ext_vector_type(2))) float v2f;
typedef __attribute__((ext_vector_type(8))) float v8f;

__device__ __forceinline__ v8f wmma_f32_4(v2f a, v2f b, v8f c) {
  // D = A(16x4) * B(4x16) + C(16x16), fp32 WMMA
  return __builtin_amdgcn_wmma_f32_16x16x4_f32(false, a, false, b, (short)0, c,
                                               false, false);
}

// ---------------------------------------------------------------- 1. KNN ----
__global__ void knn_kernel(const float* __restrict__ xyz, int* __restrict__ idx) {
  int gid = blockIdx.x * blockDim.x + threadIdx.x;   // [0, B*N)
  if (gid >= BATCH * N) return;
  int b = gid / N;
  int n = gid % N;
  const float* base = xyz + (size_t)b * N * 3;
  float qx = base[n * 3 + 0], qy = base[n * 3 + 1], qz = base[n * 3 + 2];
  float bd[KNB];
  int   bi[KNB];
#pragma unroll
  for (int s = 0; s < KNB; ++s) { bd[s] = 3.4e38f; bi[s] = 0; }
  for (int j = 0; j < N; ++j) {
    float dx = base[j * 3 + 0] - qx;
    float dy = base[j * 3 + 1] - qy;
    float dz = base[j * 3 + 2] - qz;
    float d2 = dx * dx + dy * dy + dz * dz;
    if (d2 < bd[KNB - 1]) {
#pragma unroll
      for (int s = KNB - 1; s > 0; --s) {
        if (d2 < bd[s]) {
          if (d2 >= bd[s - 1]) { bd[s] = d2; bi[s] = j; }
          else                 { bd[s] = bd[s - 1]; bi[s] = bi[s - 1]; }
        }
      }
      if (d2 < bd[0]) { bd[0] = d2; bi[0] = j; }
    }
  }
#pragma unroll
  for (int s = 0; s < KNB; ++s) idx[(size_t)gid * KNB + s] = bi[s];
}

// ------------------------------------------- 2. features + layer1 -> h -----
__global__ void feat_h_kernel(const float* __restrict__ xyz,
                              const int*   __restrict__ idx,
                              const float* __restrict__ w1,   // [16][40]
                              const float* __restrict__ b1,   // [16]
                              float*       __restrict__ h) {  // [B*N*K][16]
  int gid = blockIdx.x * blockDim.x + threadIdx.x;
  if (gid >= BATCH * N) return;
  int b = gid / N;
  int n = gid % N;
  const float* base = xyz + (size_t)b * N * 3;
  float qx = base[n * 3 + 0], qy = base[n * 3 + 1], qz = base[n * 3 + 2];

  float rel[KNB][3];
#pragma unroll
  for (int j = 0; j < KNB; ++j) {
    int id = idx[(size_t)gid * KNB + j];
    rel[j][0] = base[id * 3 + 0] - qx;
    rel[j][1] = base[id * 3 + 1] - qy;
    rel[j][2] = base[id * 3 + 2] - qz;
  }

  // covariance (uncentered, /(k-1)), sums for stats
  float cxx = 0, cxy = 0, cxz = 0, cyy = 0, cyz = 0, czz = 0;
  float mx = 0, my = 0, mz = 0, sx = 0, sy = 0, sz = 0;
  float zmax = -3.4e38f, zmin = 3.4e38f;
#pragma unroll
  for (int j = 0; j < KNB; ++j) {
    float rx = rel[j][0], ry = rel[j][1], rz = rel[j][2];
    cxx += rx * rx; cxy += rx * ry; cxz += rx * rz;
    cyy += ry * ry; cyz += ry * rz; czz += rz * rz;
    mx += rx; my += ry; mz += rz;
    float nr = sqrtf(rx * rx + ry * ry + rz * rz) + EPSF;
    sx += rx / nr; sy += ry / nr; sz += rz / nr;
    zmax = fmaxf(zmax, rz); zmin = fminf(zmin, rz);
  }
  float s2x = cxx, s2y = cyy, s2z = czz;        // sum of squares per dim
  const float inv15 = 1.0f / 15.0f;
  cxx *= inv15; cxy *= inv15; cxz *= inv15;
  cyy *= inv15; cyz *= inv15; czz *= inv15;
  mx *= (1.0f / 16.0f); my *= (1.0f / 16.0f); mz *= (1.0f / 16.0f);

  // analytic eigenvalues of symmetric 3x3 (ascending e0<=e1<=e2)
  float q  = (cxx + cyy + czz) * (1.0f / 3.0f);
  float p1 = cxy * cxy + cxz * cxz + cyz * cyz;
  float p2 = (cxx - q) * (cxx - q) + (cyy - q) * (cyy - q) + (czz - q) * (czz - q) + 2.0f * p1;
  float p  = sqrtf(fmaxf(p2, 0.0f) * (1.0f / 6.0f));
  float e0, e1, e2;
  if (p < 1e-12f) { e0 = e1 = e2 = q; }
  else {
    float ip  = 1.0f / p;
    float b00 = (cxx - q) * ip, b11 = (cyy - q) * ip, b22 = (czz - q) * ip;
    float b01 = cxy * ip, b02 = cxz * ip, b12 = cyz * ip;
    float detB = b00 * (b11 * b22 - b12 * b12) - b01 * (b01 * b22 - b12 * b02) +
                 b02 * (b01 * b12 - b11 * b02);
    float r = fminf(1.0f, fmaxf(-1.0f, 0.5f * detB));
    float phi = acosf(r) * (1.0f / 3.0f);
    e2 = q + 2.0f * p * __cosf(phi);                  // largest
    e0 = q + 2.0f * p * __cosf(phi + 2.0943951f);     // smallest
    e1 = 3.0f * q - e2 - e0;
  }
  float inv_e0 = 1.0f / (e0 + EPSF);
  float lin = (e0 - e1) * inv_e0;
  float pla = (e1 - e2) * inv_e0;
  float sph = e2 * inv_e0;

  // distance stats about the centroid
  float dsum = 0, dsq = 0, dmax = 0;
#pragma unroll
  for (int j = 0; j < KNB; ++j) {
    float dx = rel[j][0] - mx, dy = rel[j][1] - my, dz = rel[j][2] - mz;
    float d = sqrtf(dx * dx + dy * dy + dz * dz);
    dsum += d; dsq += d * d; dmax = fmaxf(dmax, d);
  }
  float dmean = dsum * (1.0f / 16.0f);
  float dstd  = sqrtf(fmaxf((dsq - 16.0f * dmean * dmean) * inv15, 0.0f));
  float dir   = (sx * sx + sy * sy + sz * sz) * (1.0f / 256.0f);
  float zstd  = sqrtf(fmaxf((s2z - 16.0f * mz * mz) * inv15, 0.0f));
  float zrng  = zmax - zmin;
  float vx = fmaxf((s2x - 16.0f * mx * mx) * inv15, 0.0f);
  float vy = fmaxf((s2y - 16.0f * my * my) * inv15, 0.0f);
  float vz = fmaxf((s2z - 16.0f * mz * mz) * inv15, 0.0f);
  float stdn = sqrtf(vx + vy + vz);

  // absolute positional encoding on grid-snapped coords
  float gx = floorf(qx), gy = floorf(qy), gz = floorf(qz);
  float ab[24];
#pragma unroll
  for (int f = 0; f < 4; ++f) {
    float fr = (float)(1 << f);
    ab[6 * f + 0] = __sinf(gx * fr); ab[6 * f + 1] = __sinf(gy * fr);
    ab[6 * f + 2] = __sinf(gz * fr); ab[6 * f + 3] = __cosf(gx * fr);
    ab[6 * f + 4] = __cosf(gy * fr); ab[6 * f + 5] = __cosf(gz * fr);
  }
  float st[13] = {lin, pla, sph, dmax, dmean, dstd, dir, zstd, zrng, mx, my, mz, stdn};

  // layer 1: only rel (cols 24..26) varies across k -> hoist the rest
  float hb[CH];
#pragma unroll
  for (int c = 0; c < CH; ++c) {
    const float* wr = w1 + c * 40;
    float acc = b1[c];
#pragma unroll
    for (int i = 0; i < 24; ++i) acc += wr[i] * ab[i];
#pragma unroll
    for (int i = 0; i < 13; ++i) acc += wr[27 + i] * st[i];
    hb[c] = acc;
  }
  float* hp = h + (size_t)gid * KNB * CH;
#pragma unroll
  for (int j = 0; j < KNB; ++j) {
#pragma unroll
    for (int c = 0; c < CH; ++c) {
      const float* wr = w1 + c * 40;
      hp[j * CH + c] = hb[c] + wr[24] * rel[j][0] + wr[25] * rel[j][1] + wr[26] * rel[j][2];
    }
  }
}

// --------------------------------------------- 3. BN stats for h (16 ch) ---
__global__ void hstats_kernel(const float* __restrict__ h,
                              const float* __restrict__ g1,
                              const float* __restrict__ be1,
                              float* __restrict__ hscale,
                              float* __restrict__ hshift) {
  const int c = blockIdx.x;          // 0..15
  const int tid = threadIdx.x;       // 256 threads
  const int CNT = BATCH * N * KNB;   // 262144
  float s = 0.0f, sq = 0.0f;
  for (int i = tid; i < CNT; i += 256) {
    float v = h[(size_t)i * CH + c];
    s += v; sq += v * v;
  }
  __shared__ float ss[256], sqs[256];
  ss[tid] = s; sqs[tid] = sq;
  __syncthreads();
  for (int o = 128; o > 0; o >>= 1) {
    if (tid < o) { ss[tid] += ss[tid + o]; sqs[tid] += sqs[tid + o]; }
    __syncthreads();
  }
  if (tid == 0) {
    float mu  = ss[0] / (float)CNT;
    float var = fmaxf(sqs[0] / (float)CNT - mu * mu, 0.0f);
    float sc  = rsqrtf(var + BN_EPS) * g1[c];
    hscale[c] = sc;
    hshift[c] = be1[c] - mu * sc;
  }
}

// -------------------------------------- 4. BN+ReLU, layer2, max_k -> pos ---
__global__ void pos_kernel(const float* __restrict__ h,
                           const float* __restrict__ hscale,
                           const float* __restrict__ hshift,
                           const float* __restrict__ w2,   // [16][16]
                           const float* __restrict__ b2,   // [16]
                           float* __restrict__ pos) {      // [B][16][N]
  int gid = blockIdx.x * blockDim.x + threadIdx.x;
  if (gid >= BATCH * N) return;
  int b = gid / N, n = gid % N;
  float sc[CH], sh[CH], best[CH];
#pragma unroll
  for (int c = 0; c < CH; ++c) { sc[c] = hscale[c]; sh[c] = hshift[c]; best[c] = -3.4e38f; }
  const float* hp = h + (size_t)gid * KNB * CH;
  for (int j = 0; j < KNB; ++j) {
    float a[CH];
#pragma unroll
    for (int c = 0; c < CH; ++c) a[c] = fmaxf(hp[j * CH + c] * sc[c] + sh[c], 0.0f);
#pragma unroll
    for (int p = 0; p < CH; ++p) {
      float acc = 0.0f;
#pragma unroll
      for (int c = 0; c < CH; ++c) acc += w2[p * CH + c] * a[c];
      best[p] = fmaxf(best[p], acc);
    }
  }
#pragma unroll
  for (int p = 0; p < CH; ++p)
    pos[((size_t)b * CH + p) * N + n] = best[p] + b2[p];
}

// ------------------------- 5. GEMM1 (WMMA): y = W1 @ [x;pos] + B1 ----------
// Each wave: 16 output channels x 64 points (4 accumulators, A reused 4x).
__global__ void gemm1_kernel(const float* __restrict__ x,    // [B,128,N]
                             const float* __restrict__ pos,  // [B,16,N]
                             const float* __restrict__ W1,   // [128,144]
                             const float* __restrict__ B1,   // [128]
                             float* __restrict__ y) {        // [B,128,N]
  const int NG = N / 64;                  // 128 groups of 4 n-tiles
  const int OT = C / 16;                  // 8
  const int TILES = BATCH * OT * NG;      // 2048 waves
  int t = blockIdx.x * (blockDim.x >> 5) + (threadIdx.x >> 5);
  if (t >= TILES) return;
  int lane = threadIdx.x & 31;
  int hi = lane >> 4, lo = lane & 15;
  int b   = t / (OT * NG);
  int rem = t % (OT * NG);
  int ot  = rem / NG, ng = rem % NG;
  int o  = ot * 16 + lo;                  // A row (M = lane%16)
  int n0 = ng * 64 + lo;                  // first B col (N = lane%16)

  const float* xb = x   + (size_t)b * C  * N;
  const float* pb = pos + (size_t)b * CH * N;

  v8f acc[4];
#pragma unroll
  for (int tt = 0; tt < 4; ++tt) acc[tt] = (v8f){0.f,0.f,0.f,0.f,0.f,0.f,0.f,0.f};

  // -- K in [0,128): B operand from x ------------------------------------
  const float* wp = W1 + o * (C + CH) + 2 * hi;   // K = v + 2*(lane/16)
  const float* xp = xb + (size_t)(2 * hi) * N + n0;
#pragma unroll 4
  for (int k0 = 0; k0 < C; k0 += 4) {
    v2f a = *(const v2f*)wp;
#pragma unroll
    for (int tt = 0; tt < 4; ++tt) {
      v2f bb;
      bb.x = xp[tt * 16];
      bb.y = xp[(size_t)N + tt * 16];
      acc[tt] = wmma_f32_4(a, bb, acc[tt]);
    }
    wp += 4;
    xp += (size_t)4 * N;
  }
  // -- K in [128,144): B operand from pos --------------------------------
  const float* pp = pb + (size_t)(2 * hi) * N + n0;
#pragma unroll
  for (int k0 = 0; k0 < CH; k0 += 4) {
    v2f a = *(const v2f*)wp;
#pragma unroll
    for (int tt = 0; tt < 4; ++tt) {
      v2f bb;
      bb.x = pp[tt * 16];
      bb.y = pp[(size_t)N + tt * 16];
      acc[tt] = wmma_f32_4(a, bb, acc[tt]);
    }
    wp += 4;
    pp += (size_t)4 * N;
  }
  // -- store: VGPR r holds row (r + 8*hi), col lane%16 --------------------
#pragma unroll
  for (int r = 0; r < 8; ++r) {
    int oo = ot * 16 + r + 8 * hi;
    float bias = B1[oo];
    float* yrow = y + ((size_t)b * C + oo) * N + n0;
#pragma unroll
    for (int tt = 0; tt < 4; ++tt) yrow[tt * 16] = acc[tt][r] + bias;
  }
}

// --------------------------------------------- 6. BN stats for y (128 ch) --
__global__ void ystats_kernel(const float* __restrict__ y,
                              const float* __restrict__ g2,
                              const float* __restrict__ be2,
                              float* __restrict__ yscale,
                              float* __restrict__ yshift) {
  const int c = blockIdx.x;        // 0..127
  const int tid = threadIdx.x;     // 256 threads
  float s = 0.0f, sq = 0.0f;
  for (int b = 0; b < BATCH; ++b) {
    const float* row = y + ((size_t)b * C + c) * N;
    for (int i = tid; i < N; i += 256) {
      float v = row[i];
      s += v; sq += v * v;
    }
  }
  __shared__ float ss[256], sqs[256];
  ss[tid] = s; sqs[tid] = sq;
  __syncthreads();
  for (int o = 128; o > 0; o >>= 1) {
    if (tid < o) { ss[tid] += ss[tid + o]; sqs[tid] += sqs[tid + o]; }
    __syncthreads();
  }
  if (tid == 0) {
    const float CNT = (float)(BATCH * N);
    float mu  = ss[0] / CNT;
    float var = fmaxf(sqs[0] / CNT - mu * mu, 0.0f);
    float sc  = rsqrtf(var + BN_EPS) * g2[c];
    yscale[c] = sc;
    yshift[c] = be2[c] - mu * sc;
  }
}

// ------------------- 7. GEMM2 (WMMA): out = W2 @ relu(bn(y)) + B2 ----------
__global__ void gemm2_kernel(const float* __restrict__ y,      // [B,128,N]
                             const float* __restrict__ yscale,
                             const float* __restrict__ yshift,
                             const float* __restrict__ W2,     // [128,128]
                             const float* __restrict__ B2,     // [128]
                             float* __restrict__ out) {        // [B,128,N]
  const int NG = N / 64;
  const int OT = C / 16;
  const int TILES = BATCH * OT * NG;      // 2048 waves
  int t = blockIdx.x * (blockDim.x >> 5) + (threadIdx.x >> 5);
  if (t >= TILES) return;
  int lane = threadIdx.x & 31;
  int hi = lane >> 4, lo = lane & 15;
  int b   = t / (OT * NG);
  int rem = t % (OT * NG);
  int ot  = rem / NG, ng = rem % NG;
  int o  = ot * 16 + lo;
  int n0 = ng * 64 + lo;

  const float* yb = y + (size_t)b * C * N;

  v8f acc[4];
#pragma unroll
  for (int tt = 0; tt < 4; ++tt) acc[tt] = (v8f){0.f,0.f,0.f,0.f,0.f,0.f,0.f,0.f};

  const float* wp = W2 + o * C + 2 * hi;
  const float* yp = yb + (size_t)(2 * hi) * N + n0;
  const float* sp = yscale + 2 * hi;
  const float* hp = yshift + 2 * hi;
#pragma unroll 4
  for (int k0 = 0; k0 < C; k0 += 4) {
    v2f a = *(const v2f*)wp;
    float s0 = sp[0], s1 = sp[1];
    float t0 = hp[0], t1 = hp[1];
#pragma unroll
    for (int tt = 0; tt < 4; ++tt) {
      v2f bb;
      bb.x = fmaxf(yp[tt * 16] * s0 + t0, 0.0f);
      bb.y = fmaxf(yp[(size_t)N + tt * 16] * s1 + t1, 0.0f);
      acc[tt] = wmma_f32_4(a, bb, acc[tt]);
    }
    wp += 4;
    yp += (size_t)4 * N;
    sp += 4;
    hp += 4;
  }
#pragma unroll
  for (int r = 0; r < 8; ++r) {
    int oo = ot * 16 + r + 8 * hi;
    float bias = B2[oo];
    float* orow = out + ((size_t)b * C + oo) * N + n0;
#pragma unroll
    for (int tt = 0; tt < 4; ++tt) orow[tt * 16] = acc[tt][r] + bias;
  }
}

} // anonymous namespace

// ---------------------------------------------------------------------------
extern "C" void kernel_launch(void* const* d_in, const int* in_sizes, int n_in,
                              void* d_out, int out_size, void* d_ws, size_t ws_size,
                              hipStream_t stream) {
  const float* x   = (const float*)d_in[0];   // [B,128,N]
  const float* xyz = (const float*)d_in[1];   // [B,N,3]
  const float* w1  = (const float*)d_in[2];   // [16,40]
  const float* b1  = (const float*)d_in[3];
  const float* g1  = (const float*)d_in[4];
  const float* be1 = (const float*)d_in[5];
  const float* w2  = (const float*)d_in[6];   // [16,16]
  const float* b2  = (const float*)d_in[7];
  const float* W1  = (const float*)d_in[8];   // [128,144]
  const float* B1  = (const float*)d_in[9];
  const float* g2  = (const float*)d_in[10];
  const float* be2 = (const float*)d_in[11];
  const float* W2  = (const float*)d_in[12];  // [128,128]
  const float* B2  = (const float*)d_in[13];
  float* out = (float*)d_out;

  // workspace layout
  char* ws = (char*)d_ws;
  int*   idx    = (int*)  (ws + 0);                       // 1 MB
  float* h      = (float*)(ws + (size_t)0x00100000);      // 16 MB
  float* pos    = (float*)(ws + (size_t)0x01100000);      // 1 MB
  float* y      = (float*)(ws + (size_t)0x01200000);      // 8 MB
  float* hscale = (float*)(ws + (size_t)0x01A00000);
  float* hshift = hscale + 16;
  float* yscale = hshift + 16;
  float* yshift = yscale + 128;

  const int PT_BLOCKS = (BATCH * N + 255) / 256;          // 64
  knn_kernel<<<PT_BLOCKS, 256, 0, stream>>>(xyz, idx);
  feat_h_kernel<<<PT_BLOCKS, 256, 0, stream>>>(xyz, idx, w1, b1, h);
  hstats_kernel<<<16, 256, 0, stream>>>(h, g1, be1, hscale, hshift);
  pos_kernel<<<PT_BLOCKS, 256, 0, stream>>>(h, hscale, hshift, w2, b2, pos);

  const int TILES = BATCH * (C / 16) * (N / 64);          // 2048 wave-tiles
  const int GB = TILES / 8;                               // 8 waves / block
  gemm1_kernel<<<GB, 256, 0, stream>>>(x, pos, W1, B1, y);
  ystats_kernel<<<128, 256, 0, stream>>>(y, g2, be2, yscale, yshift);
  gemm2_kernel<<<GB, 256, 0, stream>>>(y, yscale, yshift, W2, B2, out);
}